// RotatedDTBLLoss_8967891714142
// MI455X (gfx1250) — compile-verified
//
#include <hip/hip_runtime.h>
#include <hip/hip_bf16.h>

typedef __attribute__((ext_vector_type(2))) float v2f;
typedef __attribute__((ext_vector_type(8))) float v8f;

#define CENTER 768.0f
#define LOGCLAMP -100.0f

struct SelState {
    unsigned int hist[256];
    unsigned int prefix;
    unsigned int kprime;
    unsigned int tau;
    unsigned int pad0;
    double score_sum;   // sum of t_scores
    double fg_sum;      // sum of top-k scores
    double qfl_sum;     // sum of per-element losses
    double cnt_gt;      // count of scores strictly > tau
};

// ---------------- reductions (deterministic) ----------------
__device__ __forceinline__ double warpSumD(double v) {
    for (int off = 16; off; off >>= 1) v += __shfl_xor(v, off, 32);
    return v;
}
__device__ double blockSumD(double v) {
    __shared__ double sh[8];
    int lane = threadIdx.x & 31, w = threadIdx.x >> 5;
    v = warpSumD(v);
    if (lane == 0) sh[w] = v;
    __syncthreads();
    double r = 0.0;
    if (w == 0) {
        r = (lane < (int)(blockDim.x >> 5)) ? sh[lane] : 0.0;
        r = warpSumD(r);
    }
    __syncthreads();
    return r;  // valid in thread 0
}

__device__ __forceinline__ float sigm(float x) {
    return 1.0f / (1.0f + expf(-x));
}

// ---------------- init ----------------
__global__ void init_kernel(SelState* st, int k) {
    if (threadIdx.x < 256) st->hist[threadIdx.x] = 0;
    if (threadIdx.x == 0) {
        st->prefix = 0u;
        st->kprime = (unsigned)k;
        st->tau = 0u;
        st->score_sum = 0.0;
        st->fg_sum = 0.0;
        st->qfl_sum = 0.0;
        st->cnt_gt = 0.0;
    }
}

// ---------------- t_scores + partial sums ----------------
__global__ void score_kernel(const float* __restrict__ t_cls, float* __restrict__ scores,
                             double* __restrict__ part, int N, int C) {
    int i = blockIdx.x * blockDim.x + threadIdx.x;
    double local = 0.0;
    if (i < N) {
        float m = -3.4e38f;
        const float* row = t_cls + (long long)i * C;
        for (int c = 0; c < C; ++c) m = fmaxf(m, row[c]);
        float s = sigm(m);
        scores[i] = s;
        local = (double)s;
    }
    double b = blockSumD(local);
    if (threadIdx.x == 0) part[blockIdx.x] = b;
}

// ---------------- radix select: histogram pass ----------------
__global__ void hist_kernel(const float* __restrict__ scores, SelState* st, int N, int d) {
    __shared__ unsigned int h[256];
    h[threadIdx.x] = 0u;
    __syncthreads();
    int i = blockIdx.x * blockDim.x + threadIdx.x;
    if (i < N) {
        unsigned int v = __float_as_uint(scores[i]);
        bool ok;
        if (d == 0) {
            ok = true;
        } else {
            unsigned int sh = (unsigned)(32 - 8 * d) & 31u;
            ok = ((v >> sh) == st->prefix);
        }
        if (ok) {
            unsigned int b = (v >> (unsigned)(24 - 8 * d)) & 255u;
            atomicAdd(&h[b], 1u);
        }
    }
    __syncthreads();
    if (h[threadIdx.x]) atomicAdd(&st->hist[threadIdx.x], h[threadIdx.x]);
}

// ---------------- radix select: scan (1 thread) ----------------
__global__ void scan_kernel(SelState* st, int d) {
    unsigned int kp = st->kprime;
    unsigned int cum = 0, sel = 0;
    for (int b = 255; b >= 0; --b) {
        unsigned int c = st->hist[b];
        if (cum + c >= kp) { sel = (unsigned)b; kp -= cum; break; }
        cum += c;
    }
    st->prefix = (st->prefix << 8) | sel;
    st->kprime = kp;
    for (int b = 0; b < 256; ++b) st->hist[b] = 0u;
    if (d == 3) st->tau = st->prefix;
}

// ---------------- mark strictly-greater, partial count & sum ----------------
__global__ void mark_kernel(const float* __restrict__ scores, unsigned char* __restrict__ mask,
                            const SelState* st, double* __restrict__ cntPart,
                            double* __restrict__ fgPart, int N) {
    unsigned int tau = st->tau;
    int i = blockIdx.x * blockDim.x + threadIdx.x;
    double lc = 0.0, ls = 0.0;
    if (i < N) {
        float s = scores[i];
        unsigned int v = __float_as_uint(s);
        bool gt = v > tau;
        mask[i] = gt ? 1 : 0;
        if (gt) { lc = 1.0; ls = (double)s; }
    }
    double bc = blockSumD(lc);
    double bs = blockSumD(ls);
    if (threadIdx.x == 0) { cntPart[blockIdx.x] = bc; fgPart[blockIdx.x] = bs; }
}

// ---------------- fixed-order reduction of partials ----------------
__global__ void reduce_kernel(const double* __restrict__ part, int n, double* __restrict__ slot) {
    double v = 0.0;
    for (int i = threadIdx.x; i < n; i += blockDim.x) v += part[i];
    v = blockSumD(v);
    if (threadIdx.x == 0) *slot = v;
}

// ---------------- deterministic tie handling (1 wave) ----------------
__global__ void tie_kernel(const float* __restrict__ scores, unsigned char* __restrict__ mask,
                           SelState* st, int N, int k) {
    unsigned int tau = st->tau;
    unsigned int cgt = (unsigned int)(st->cnt_gt + 0.5);
    unsigned int need = (unsigned)k - cgt;
    int lane = threadIdx.x;
    unsigned int taken = 0;
    for (int base = 0; base < N && taken < need; base += 32) {
        int i = base + lane;
        bool eq = (i < N) && (__float_as_uint(scores[i]) == tau);
        unsigned long long bal = __ballot(eq);
        unsigned int rank = (unsigned)__popcll(bal & ((1ull << lane) - 1ull));
        if (eq && (taken + rank) < need) mask[i] = 1;
        taken += (unsigned)__popcll(bal);
    }
    if (lane == 0) {
        st->fg_sum += (double)need * (double)__uint_as_float(tau);
    }
}

// ---------------- QFL loss ----------------
__global__ void qfl_kernel(const float* __restrict__ s_cls, const float* __restrict__ t_cls,
                           const unsigned char* __restrict__ mask, double* __restrict__ part,
                           int total, int C) {
    int i = blockIdx.x * blockDim.x + threadIdx.x;
    double local = 0.0;
    if (i < total) {
        int row = i / C;
        float p = sigm(s_cls[i]);
        float l1p = fmaxf(logf(1.0f - p), LOGCLAMP);
        float loss;
        if (mask[row]) {
            float t = sigm(t_cls[i]);
            float lp = fmaxf(logf(p), LOGCLAMP);
            float bce = -(t * lp + (1.0f - t) * l1p);
            float df = t - p;
            loss = bce * df * df;
        } else {
            loss = -l1p * p * p;
        }
        local = (double)loss;
    }
    double b = blockSumD(local);
    if (threadIdx.x == 0) part[blockIdx.x] = b;
}

// ---------------- per-box GEMM coefficients ----------------
// weighted[p,g] = dot(f[p], coef[g]),  f = [px'^2, py'^2, px'*py', px', py', 1, 0, 0]
// (coords centered at CENTER for precision)
__global__ void coef_kernel(const float* __restrict__ bb, const float* __restrict__ gsc,
                            float* __restrict__ coef, int G) {
    int g = blockIdx.x * blockDim.x + threadIdx.x;
    if (g >= G) return;
    float cx = bb[g * 5 + 0] - CENTER;
    float cy = bb[g * 5 + 1] - CENTER;
    float w  = bb[g * 5 + 2];
    float h  = bb[g * 5 + 3];
    float an = bb[g * 5 + 4];
    float ca = cosf(an), sa = sinf(an);
    float iw = 4.0f / (w * w);
    float ih = 4.0f / (h * h);
    float A  = ca * ca * iw + sa * sa * ih;
    float B  = sa * sa * iw + ca * ca * ih;
    float Cc = 2.0f * ca * sa * (iw - ih);
    float gs = gsc[g];
    coef[g * 8 + 0] = -A * gs;
    coef[g * 8 + 1] = -B * gs;
    coef[g * 8 + 2] = -Cc * gs;
    coef[g * 8 + 3] = gs * (2.0f * A * cx + Cc * cy);
    coef[g * 8 + 4] = gs * (2.0f * B * cy + Cc * cx);
    coef[g * 8 + 5] = gs * (1.0f - (A * cx * cx + B * cy * cy + Cc * cx * cy));
    coef[g * 8 + 6] = 0.0f;
    coef[g * 8 + 7] = 0.0f;
}

// ---------------- WMMA GEMM + rowmax + exact refinement ----------------
// One wave per 16-point tile. Loop over G/16 box tiles; two K=4 fp32 WMMAs per tile.
// C layout (16x16 f32): lanes 0-15 -> (M=r, N=lane) in VGPR r; lanes 16-31 -> (M=r+8, N=lane-16).
__global__ void __launch_bounds__(256) gauss_kernel(const float* __restrict__ points,
                                                    const float* __restrict__ coef,
                                                    const float* __restrict__ bb,
                                                    const float* __restrict__ gsc,
                                                    float* __restrict__ out_gc,
                                                    int P, int G) {
    int wave = threadIdx.x >> 5;
    int lane = threadIdx.x & 31;
    int tile = blockIdx.x * (blockDim.x >> 5) + wave;
    int numTiles = (P + 15) >> 4;
    if (tile >= numTiles) return;  // wave-uniform

    int M = lane & 15;
    int pi = tile * 16 + M;
    int piC = pi < P ? pi : (P - 1);
    float px = points[2 * piC + 0] - CENTER;
    float py = points[2 * piC + 1] - CENTER;

    bool lo = lane < 16;
    // A chunks: K0..3 = [px^2, py^2, px*py, px]; K4..7 = [py, 1, 0, 0]
    v2f a0, a1;
    a0.x = lo ? px * px : px * py;
    a0.y = lo ? py * py : px;
    a1.x = lo ? py : 0.0f;
    a1.y = lo ? 1.0f : 0.0f;

    float best[8];
    int bestg[8];
#pragma unroll
    for (int r = 0; r < 8; ++r) { best[r] = -3.4e38f; bestg[r] = 0; }

    int nGT = G >> 4;
    for (int jt = 0; jt < nGT; ++jt) {
        int g = jt * 16 + (lane & 15);
        const float* cf = coef + g * 8;
        v2f b0, b1;
        b0.x = lo ? cf[0] : cf[2];
        b0.y = lo ? cf[1] : cf[3];
        b1.x = lo ? cf[4] : cf[6];
        b1.y = lo ? cf[5] : cf[7];

        v8f c = {};
        c = __builtin_amdgcn_wmma_f32_16x16x4_f32(false, a0, false, b0, (short)0, c, false, false);
        c = __builtin_amdgcn_wmma_f32_16x16x4_f32(false, a1, false, b1, (short)0, c, false, false);

#pragma unroll
        for (int r = 0; r < 8; ++r) {
            float v = c[r];
            if (v > best[r]) { best[r] = v; bestg[r] = jt * 16 + (lane & 15); }
        }
    }

    // reduce (val, argmax) across the 16 lanes of each half, per row slot r
#pragma unroll
    for (int r = 0; r < 8; ++r) {
        float v = best[r];
        int g = bestg[r];
        for (int off = 8; off >= 1; off >>= 1) {
            float ov = __shfl_xor(v, off, 32);
            int og = __shfl_xor(g, off, 32);
            if (ov > v) { v = ov; g = og; }
        }
        best[r] = v;
        bestg[r] = g;
    }

    // lane 0 holds rows 0..7, lane 16 holds rows 8..15; refine exactly
    if ((lane & 15) == 0) {
        int rowOff = (lane >= 16) ? 8 : 0;
#pragma unroll
        for (int r = 0; r < 8; ++r) {
            int row = r + rowOff;
            int p = tile * 16 + row;
            if (p >= P) continue;
            int g = bestg[r];
            float cx = bb[g * 5 + 0];
            float cy = bb[g * 5 + 1];
            float w  = bb[g * 5 + 2];
            float h  = bb[g * 5 + 3];
            float an = bb[g * 5 + 4];
            float ca = cosf(an), sa = sinf(an);
            float dx = points[2 * p + 0] - cx;
            float dy = points[2 * p + 1] - cy;
            float ox = ca * dx + sa * dy;
            float oy = -sa * dx + ca * dy;
            float hw = 0.5f * w, hh = 0.5f * h;
            float gc = 1.0f - (ox * ox / (hw * hw) + oy * oy / (hh * hh));
            float wv = gc * gsc[g];
            out_gc[p] = fminf(fmaxf(wv, 0.0f), 1.0f);
        }
    }
}

// ---------------- finalize scalars ----------------
__global__ void finalize_kernel(const SelState* st, float* __restrict__ out, int N, int C, int P) {
    out[0] = (float)(st->qfl_sum / ((double)N * (double)C));
    out[P + 1] = (float)st->fg_sum;
    out[P + 2] = (float)(st->score_sum / (double)N);
}

// ---------------- launch ----------------
extern "C" void kernel_launch(void* const* d_in, const int* in_sizes, int n_in,
                              void* d_out, int out_size, void* d_ws, size_t ws_size,
                              hipStream_t stream) {
    const float* s_cls  = (const float*)d_in[0];
    const float* t_cls  = (const float*)d_in[1];
    const float* bboxes = (const float*)d_in[3];
    const float* points = (const float*)d_in[4];
    const float* gscore = (const float*)d_in[5];
    float* out = (float*)d_out;

    const int N = in_sizes[2];
    const int C = in_sizes[0] / N;
    const int P = in_sizes[4] / 2;
    const int G = in_sizes[5];
    int k = (int)((double)N * 0.01);
    if (k < 2) k = 2;

    // workspace layout
    char* ws = (char*)d_ws;
    size_t off = 0;
    float* scores = (float*)(ws + off);           off += (size_t)N * 4;
    unsigned char* mask = (unsigned char*)(ws + off); off += (size_t)N;
    off = (off + 63) & ~(size_t)63;
    SelState* st = (SelState*)(ws + off);         off += (sizeof(SelState) + 63) & ~(size_t)63;
    float* coef = (float*)(ws + off);             off += (size_t)G * 8 * 4;
    off = (off + 63) & ~(size_t)63;
    const int nb1 = (N + 255) / 256;
    const int total = N * C;
    const int nbq = (total + 255) / 256;
    double* scorePart = (double*)(ws + off);      off += (size_t)nb1 * 8;
    double* cntPart = (double*)(ws + off);        off += (size_t)nb1 * 8;
    double* fgPart = (double*)(ws + off);         off += (size_t)nb1 * 8;
    double* qflPart = (double*)(ws + off);        off += (size_t)nbq * 8;

    init_kernel<<<1, 256, 0, stream>>>(st, k);
    score_kernel<<<nb1, 256, 0, stream>>>(t_cls, scores, scorePart, N, C);
    reduce_kernel<<<1, 256, 0, stream>>>(scorePart, nb1, &st->score_sum);

    for (int d = 0; d < 4; ++d) {
        hist_kernel<<<nb1, 256, 0, stream>>>(scores, st, N, d);
        scan_kernel<<<1, 1, 0, stream>>>(st, d);
    }

    mark_kernel<<<nb1, 256, 0, stream>>>(scores, mask, st, cntPart, fgPart, N);
    reduce_kernel<<<1, 256, 0, stream>>>(cntPart, nb1, &st->cnt_gt);
    reduce_kernel<<<1, 256, 0, stream>>>(fgPart, nb1, &st->fg_sum);
    tie_kernel<<<1, 32, 0, stream>>>(scores, mask, st, N, k);

    qfl_kernel<<<nbq, 256, 0, stream>>>(s_cls, t_cls, mask, qflPart, total, C);
    reduce_kernel<<<1, 256, 0, stream>>>(qflPart, nbq, &st->qfl_sum);

    coef_kernel<<<(G + 255) / 256, 256, 0, stream>>>(bboxes, gscore, coef, G);
    {
        int numTiles = (P + 15) / 16;
        int wpb = 8;  // 256 threads = 8 waves
        int blocks = (numTiles + wpb - 1) / wpb;
        gauss_kernel<<<blocks, 256, 0, stream>>>(points, coef, bboxes, gscore, out + 1, P, G);
    }

    finalize_kernel<<<1, 1, 0, stream>>>(st, out, N, C, P);
}